// TripletLoss_16836271800774
// MI455X (gfx1250) — compile-verified
//
#include <hip/hip_runtime.h>

#define BATCH 512
#define DIMS 128
#define MARGIN_F 0.2f
#define MAXT 200

typedef float v2f __attribute__((ext_vector_type(2)));
typedef float v8f __attribute__((ext_vector_type(8)));

// ---------------------------------------------------------------------------
// Kernel 0: squared row norms  sq[i] = sum_k E[i][k]^2
// ---------------------------------------------------------------------------
__global__ void tl_sq_kernel(const float* __restrict__ E, float* __restrict__ sq) {
    int i = blockIdx.x * blockDim.x + threadIdx.x;
    if (i < BATCH) {
        const float* row = E + i * DIMS;
        float s = 0.0f;
#pragma unroll 8
        for (int k = 0; k < DIMS; ++k) s += row[k] * row[k];
        sq[i] = s;
    }
}

// ---------------------------------------------------------------------------
// Kernel 1: D[i][j] = max(sq[i] + sq[j] - 2*(E E^T)[i][j], 0)
// One wave (32 lanes) per 16x16 tile, V_WMMA_F32_16X16X4_F32 over K.
//
// f32 A-matrix (16x4) layout: lanes 0-15 -> M=lane, VGPR v -> K=v;
//                             lanes 16-31 -> M=lane-16, VGPR v -> K=v+2.
// B (4x16) tile of E^T for rows j has the identical per-lane pattern.
// C/D (16x16 f32): lane = N + 16*half, VGPR r -> M = r + 8*half.
// ---------------------------------------------------------------------------
__global__ void tl_gram_dist_kernel(const float* __restrict__ E,
                                    const float* __restrict__ sq,
                                    float* __restrict__ D) {
    const int tj   = blockIdx.x;        // column tile index
    const int ti   = blockIdx.y;        // row tile index
    const int lane = threadIdx.x;       // 0..31 (one wave)
    const int half = lane >> 4;         // 0 or 1
    const int l16  = lane & 15;

    const int i0 = ti * 16;
    const int j0 = tj * 16;

    const float* rowA = E + (i0 + l16) * DIMS + half * 2;
    const float* rowB = E + (j0 + l16) * DIMS + half * 2;

    v8f c = {};
#pragma unroll
    for (int k = 0; k < DIMS; k += 4) {
        v2f a = *(const v2f*)(rowA + k);
        v2f b = *(const v2f*)(rowB + k);
        // 8 args: (neg_a, A, neg_b, B, c_mod, C, reuse_a, reuse_b)
        c = __builtin_amdgcn_wmma_f32_16x16x4_f32(false, a, false, b,
                                                  (short)0, c, false, false);
    }

    const int   j   = j0 + l16;
    const float sqj = sq[j];
#pragma unroll
    for (int r = 0; r < 8; ++r) {
        int   i = i0 + r + half * 8;
        float d = sq[i] + sqj - 2.0f * c[r];
        D[i * BATCH + j] = fmaxf(d, 0.0f);
    }
}

// ---------------------------------------------------------------------------
// Kernel 2: per-anchor semi-hard mining.
// For each (a,p) with labels[p]==labels[a], p!=a: find FIRST n (ascending)
// with labels[n]!=labels[a] and D_ap < D_an < D_ap + margin.
// Emits mined flag + contrib per pair, and per-anchor (count, contrib-sum)
// via a deterministic shared-memory tree reduction.
// ---------------------------------------------------------------------------
__global__ void tl_mine_kernel(const float* __restrict__ D,
                               const int* __restrict__ labels,
                               unsigned char* __restrict__ mined,
                               float* __restrict__ contrib,
                               int* __restrict__ cnt,
                               float* __restrict__ sumC) {
    __shared__ float        rowD[BATCH];
    __shared__ unsigned int negMask[BATCH / 32];
    __shared__ int          scnt[256];
    __shared__ float        ssum[256];

    const int a   = blockIdx.x;
    const int tid = threadIdx.x;          // 0..255
    const int la  = labels[a];

    for (int p = tid; p < BATCH; p += 256) rowD[p] = D[a * BATCH + p];
    if (tid < BATCH / 32) {
        unsigned int m = 0;
        for (int b = 0; b < 32; ++b)
            if (labels[tid * 32 + b] != la) m |= (1u << b);
        negMask[tid] = m;
    }
    __syncthreads();

    int   lc = 0;
    float ls = 0.0f;
    for (int p = tid; p < BATCH; p += 256) {
        unsigned char mi = 0;
        float         co = 0.0f;
        if (p != a && labels[p] == la) {
            const float dap = rowD[p];
            const float hi  = dap + MARGIN_F;
            bool found = false;
            for (int w = 0; w < BATCH / 32 && !found; ++w) {
                unsigned int m = negMask[w];
                while (m) {                       // ascending n via lowest set bit
                    int   b   = __ffs(m) - 1;
                    m &= m - 1;
                    float dan = rowD[w * 32 + b];
                    if (dap < dan && dan < hi) {
                        co    = fmaxf(dap - dan + MARGIN_F, 0.0f);
                        found = true;
                        break;
                    }
                }
            }
            if (found) { mi = 1; lc += 1; ls += co; }
        }
        mined[a * BATCH + p]   = mi;
        contrib[a * BATCH + p] = co;
    }

    scnt[tid] = lc;
    ssum[tid] = ls;
    __syncthreads();
#pragma unroll
    for (int s = 128; s > 0; s >>= 1) {
        if (tid < s) { scnt[tid] += scnt[tid + s]; ssum[tid] += ssum[tid + s]; }
        __syncthreads();
    }
    if (tid == 0) { cnt[a] = scnt[0]; sumC[a] = ssum[0]; }
}

// ---------------------------------------------------------------------------
// Kernel 3: stable-sort-by-label selection of first MAXT mined triplets.
// Order: label value ascending -> anchor index ascending -> p ascending.
// Whole anchors are folded via precomputed sums; only the boundary anchor
// needs a p-scan. Serial, tiny, fully deterministic.
// ---------------------------------------------------------------------------
__global__ void tl_finalize_kernel(const int* __restrict__ labels,
                                   const int* __restrict__ cnt,
                                   const float* __restrict__ sumC,
                                   const unsigned char* __restrict__ mined,
                                   const float* __restrict__ contrib,
                                   float* __restrict__ out) {
    if (threadIdx.x != 0 || blockIdx.x != 0) return;

    int totalMined = 0;
    for (int a = 0; a < BATCH; ++a) totalMined += cnt[a];

    int   running = 0;
    float total   = 0.0f;
    bool  full    = false;
    for (int v = 0; v < 32 && !full; ++v) {
        for (int a = 0; a < BATCH; ++a) {
            if (labels[a] != v) continue;
            int c = cnt[a];
            if (running + c <= MAXT) {
                total   += sumC[a];
                running += c;
            } else {
                int take = MAXT - running;
                int got  = 0;
                for (int p = 0; p < BATCH && got < take; ++p) {
                    if (mined[a * BATCH + p]) {
                        total += contrib[a * BATCH + p];
                        ++got;
                    }
                }
                running = MAXT;
            }
            if (running >= MAXT) { full = true; break; }
        }
    }

    int count = totalMined < MAXT ? totalMined : MAXT;
    if (count < 1) count = 1;
    out[0] = total / (float)count;
}

// ---------------------------------------------------------------------------
extern "C" void kernel_launch(void* const* d_in, const int* in_sizes, int n_in,
                              void* d_out, int out_size, void* d_ws, size_t ws_size,
                              hipStream_t stream) {
    (void)in_sizes; (void)n_in; (void)out_size; (void)ws_size;

    const float* E      = (const float*)d_in[0];
    const int*   labels = (const int*)d_in[1];

    char* ws = (char*)d_ws;
    float*         D       = (float*)(ws);                          // 1 MB
    float*         contrib = (float*)(ws + 1048576);                // 1 MB
    unsigned char* mined   = (unsigned char*)(ws + 2097152);        // 256 KB
    float*         sq      = (float*)(ws + 2359296);                // 2 KB
    int*           cnt     = (int*)(ws + 2361344);                  // 2 KB
    float*         sumC    = (float*)(ws + 2363392);                // 2 KB

    tl_sq_kernel<<<2, 256, 0, stream>>>(E, sq);

    dim3 grid(BATCH / 16, BATCH / 16);                              // 32x32 tiles
    tl_gram_dist_kernel<<<grid, 32, 0, stream>>>(E, sq, D);

    tl_mine_kernel<<<BATCH, 256, 0, stream>>>(D, labels, mined, contrib, cnt, sumC);

    tl_finalize_kernel<<<1, 32, 0, stream>>>(labels, cnt, sumC, mined, contrib,
                                             (float*)d_out);
}